// G2ITranslator_46273977647530
// MI455X (gfx1250) — compile-verified
//
#include <hip/hip_runtime.h>
#include <math.h>

typedef __attribute__((ext_vector_type(2))) float v2f;
typedef __attribute__((ext_vector_type(8))) float v8f;

#define NPAIR 120   // valid pairs per image (16 choose 2)
#define NIMG  24
#define H     128

// ---------------------------------------------------------------------------
// Kernel 1: per-pair parameter setup.
// For pair p (image b, nodes i<j):
//   ix = (x - 127*x0)/(x1-x0) = sx*x + bx   (pixel-space sampling coordinate)
//   t  = clip(2*sigmoid(0.03*(dual_slope-2))-1, 0, 1)
// Stores SoA arrays: sx, bx, sy, by, vec*t, vec*(1-t).
// ---------------------------------------------------------------------------
__global__ void pair_setup_kernel(const float* __restrict__ nodes,
                                  const float* __restrict__ adj,
                                  float* __restrict__ P) {
    int p = blockIdx.x * blockDim.x + threadIdx.x;
    const int N = NIMG * NPAIR;
    if (p >= N) return;
    int b = p / NPAIR;
    int q = p - b * NPAIR;
    // decode lexicographic (i,j), i<j, from q
    int i = 0, rem = q;
    while (rem >= 15 - i) { rem -= 15 - i; ++i; }
    int j = i + 1 + rem;

    int na = b * 16 + i, nbn = b * 16 + j;
    float x0 = nodes[2 * na],     y0 = nodes[2 * na + 1];
    float x1 = nodes[2 * nbn],    y1 = nodes[2 * nbn + 1];
    float vec = adj[b * 256 + i * 16 + j];

    float dx = x1 - x0, dy = y1 - y0;
    float ds = fabsf(dy / (fabsf(dx) + 1e-6f)) + fabsf(dx / (fabsf(dy) + 1e-6f));
    float z = 0.03f * (ds - 2.0f);
    float sg = 1.0f / (1.0f + __expf(-z));
    float t = fminf(fmaxf(2.0f * sg - 1.0f, 0.0f), 1.0f);

    float sx = 1.0f / dx;
    float bx = -127.0f * x0 * sx;
    float sy = 1.0f / dy;
    float by = -127.0f * y0 * sy;

    P[0 * N + p] = sx;
    P[1 * N + p] = bx;
    P[2 * N + p] = sy;
    P[3 * N + p] = by;
    P[4 * N + p] = vec * t;            // coef for separable (WMMA) part
    P[5 * N + p] = vec * (1.0f - t);   // coef for band part
}

// Sum of in-bounds bilinear weights along one axis (the "marginal").
__device__ __forceinline__ float axis_marginal(float s, float b, float coord) {
    float ix = fmaf(s, coord, b);
    float f = floorf(ix);
    float w = ix - f;
    float W0 = (f >= 0.0f && f <= 127.0f) ? (1.0f - w) : 0.0f;
    float W1 = (f >= -1.0f && f <= 126.0f) ? w : 0.0f;
    return W0 + W1;
}

// ---------------------------------------------------------------------------
// Kernel 2: one wave per 16x16 output tile.
//  Part 1 (WMMA): Out_t(y,x) = sum_p (vec_p*t_p) * Ay_p(y) * Ax_p(x)
//    = 128x128x120 GEMM per image via v_wmma_f32_16x16x4_f32 (30 k-steps),
//    A/B fragments synthesized on the fly from per-pair slope/intercept.
//  Part 2 (VALU): band term sum_p (vec_p*(1-t_p)) * g_p(y,x) accumulated into
//    the same v8f accumulator (WMMA C/D layout: VGPR r -> rows r / r+8).
//  Then clip to [0,1] and store.
// ---------------------------------------------------------------------------
__global__ __launch_bounds__(256) void layout_kernel(const float* __restrict__ P,
                                                     float* __restrict__ out) {
    __shared__ float sSX[NPAIR], sBX[NPAIR], sSY[NPAIR], sBY[NPAIR],
                     sCT[NPAIR], sCL[NPAIR];
    const int N = NIMG * NPAIR;

    int b     = blockIdx.x >> 3;        // 8 blocks per image
    int tbase = (blockIdx.x & 7) * 8;   // first tile handled by this block
    int lane  = threadIdx.x & 31;
    int wave  = threadIdx.x >> 5;

    for (int idx = threadIdx.x; idx < NPAIR; idx += blockDim.x) {
        int p = b * NPAIR + idx;
        sSX[idx] = P[0 * N + p];
        sBX[idx] = P[1 * N + p];
        sSY[idx] = P[2 * N + p];
        sBY[idx] = P[3 * N + p];
        sCT[idx] = P[4 * N + p];
        sCL[idx] = P[5 * N + p];
    }
    __syncthreads();

    int tile = tbase + wave;            // 0..63 within image
    int ty = tile >> 3, tx = tile & 7;
    int m  = lane & 15;
    int hi = lane >> 4;

    float yA = (float)(ty * 16 + m);    // A-fragment row coordinate
    float xB = (float)(tx * 16 + m);    // B-fragment col coordinate

    v8f acc = {0.f, 0.f, 0.f, 0.f, 0.f, 0.f, 0.f, 0.f};

    // ---- separable part: GEMM over the 120 pairs, K=4 per WMMA ----
    #pragma unroll 2
    for (int k0 = 0; k0 < NPAIR; k0 += 4) {
        int kk = k0 + hi * 2;           // this half-wave's K pair indices
        v2f a, bv;
        a.x  = sCT[kk]     * axis_marginal(sSY[kk],     sBY[kk],     yA);
        a.y  = sCT[kk + 1] * axis_marginal(sSY[kk + 1], sBY[kk + 1], yA);
        bv.x = axis_marginal(sSX[kk],     sBX[kk],     xB);
        bv.y = axis_marginal(sSX[kk + 1], sBX[kk + 1], xB);
        acc = __builtin_amdgcn_wmma_f32_16x16x4_f32(
            /*neg_a=*/false, a, /*neg_b=*/false, bv,
            /*c_mod=*/(short)0, acc, /*reuse_a=*/false, /*reuse_b=*/false);
    }

    // ---- band part: 3 indicators on d = floor(iy)-floor(ix) per pair ----
    float xg    = xB;                          // this lane's output x
    float ybase = (float)(ty * 16 + hi * 8);   // rows ybase..ybase+7
    #pragma unroll 1
    for (int p = 0; p < NPAIR; ++p) {
        float sx = sSX[p], bx = sBX[p], sy = sSY[p], by = sBY[p], cl = sCL[p];
        float ix = fmaf(sx, xg, bx);
        float fx = floorf(ix);
        float wx = ix - fx;
        float Wx0 = (fx >= 0.0f && fx <= 127.0f) ? (1.0f - wx) : 0.0f;
        float Wx1 = (fx >= -1.0f && fx <= 126.0f) ? wx : 0.0f;
        float iy0 = fmaf(sy, ybase, by);
        #pragma unroll
        for (int r = 0; r < 8; ++r) {
            float iy = iy0 + sy * (float)r;
            float fy = floorf(iy);
            float wy = iy - fy;
            float Wy0 = (fy >= 0.0f && fy <= 127.0f) ? (1.0f - wy) : 0.0f;
            float Wy1 = (fy >= -1.0f && fy <= 126.0f) ? wy : 0.0f;
            float d  = fy - fx;
            float I0 = (fabsf(d)        <= 10.0f) ? 1.0f : 0.0f; // (yc0,xc0)&(yc1,xc1)
            float Im = (fabsf(d - 1.0f) <= 10.0f) ? 1.0f : 0.0f; // (yc0,xc1)
            float Ip = (fabsf(d + 1.0f) <= 10.0f) ? 1.0f : 0.0f; // (yc1,xc0)
            float g = (Wy0 * Wx0 + Wy1 * Wx1) * I0 + Wy0 * Wx1 * Im + Wy1 * Wx0 * Ip;
            acc[r] += cl * g;
        }
    }

    // ---- clip & store (C/D layout: VGPR r -> row r (lanes 0-15) / r+8) ----
    float* op = out + b * (H * H);
    int x = tx * 16 + (lane & 15);
    #pragma unroll
    for (int r = 0; r < 8; ++r) {
        int y = ty * 16 + hi * 8 + r;
        op[y * H + x] = fminf(fmaxf(acc[r], 0.0f), 1.0f);
    }
}

extern "C" void kernel_launch(void* const* d_in, const int* in_sizes, int n_in,
                              void* d_out, int out_size, void* d_ws, size_t ws_size,
                              hipStream_t stream) {
    const float* nodes = (const float*)d_in[0];   // (384, 2) f32
    const float* adj   = (const float*)d_in[1];   // (6144, 1) f32
    // d_in[2] (nodes_per_image) is a compile-time constant (24 x 16) -> unused.
    float* params = (float*)d_ws;                 // 6 * 2880 floats = 69 KB
    float* out    = (float*)d_out;                // 24*1*128*128 f32

    pair_setup_kernel<<<(NIMG * NPAIR + 255) / 256, 256, 0, stream>>>(nodes, adj, params);
    layout_kernel<<<NIMG * 8, 256, 0, stream>>>(params, out);
}